// LowLightApdaptiveCNNBlock_33277406610154
// MI455X (gfx1250) — compile-verified
//
#include <hip/hip_runtime.h>
#include <math.h>

#define EPS   1e-5f
#define B_    16
#define C_    256
#define H_    96
#define W_    96
#define E_    8
#define HW_   (H_ * W_)
#define NKJ   (C_ * 9)        // 2304, flattened (c, ky, kx)
#define TW    (W_ + 2)        // 98

typedef float v2f __attribute__((ext_vector_type(2)));
typedef float v8f __attribute__((ext_vector_type(8)));

#if __has_builtin(__builtin_amdgcn_global_load_async_to_lds_b32)
#define HAVE_ASYNC_LDS 1
typedef __attribute__((address_space(1))) int gl_i32;   // global (printed __device__)
typedef __attribute__((address_space(3))) int lds_i32;  // LDS
#endif

// ---------------------------------------------------------------------------
// Kernel A: per-(b,c) plane statistics over x (vectorized float4 reads).
// stats[bc*9 + {0..8}] = {total, rowsum0, rowsum95, colsum0, colsum95,
//                         x[0][0], x[0][95], x[95][0], x[95][95]}
// ---------------------------------------------------------------------------
__global__ void stats_kernel(const float* __restrict__ x, float* __restrict__ stats) {
    __shared__ float sred[256];
    const int bc = blockIdx.x;            // 0..4095 == b*256 + c
    const int t  = threadIdx.x;           // 0..255
    const float* p = x + (size_t)bc * HW_;
    const float4* p4 = (const float4*)p;

    float tot = 0.f, rs0 = 0.f, rsN = 0.f, cs0 = 0.f, csN = 0.f;
    for (int i = 0; i < HW_ / (4 * 256); ++i) {     // 9 iterations
        int q = t + i * 256;                        // float4 index, 0..2303
        float4 v = p4[q];
        int h  = q / 24;                            // 24 float4 per row
        int ws = (q - h * 24) * 4;
        float s = v.x + v.y + v.z + v.w;
        tot += s;
        if (h == 0)       rs0 += s;
        if (h == H_ - 1)  rsN += s;
        if (ws == 0)      cs0 += v.x;
        if (ws == W_ - 4) csN += v.w;
    }

    float vals[5] = {tot, rs0, rsN, cs0, csN};
    for (int k = 0; k < 5; ++k) {
        sred[t] = vals[k];
        __syncthreads();
        for (int s = 128; s > 0; s >>= 1) {
            if (t < s) sred[t] += sred[t + s];
            __syncthreads();
        }
        if (t == 0) stats[bc * 9 + k] = sred[0];
        __syncthreads();
    }
    if (t == 0) {
        stats[bc * 9 + 5] = p[0];                 // x[0][0]
        stats[bc * 9 + 6] = p[W_ - 1];            // x[0][95]
        stats[bc * 9 + 7] = p[(H_ - 1) * W_];     // x[95][0]
        stats[bc * 9 + 8] = p[HW_ - 1];           // x[95][95]
    }
}

// ---------------------------------------------------------------------------
// Kernel B: routing activations r[b][c] = relu(BN(mean(dwconv)+bias)) from
// stats via inclusion-exclusion (routing conv collapsed analytically).
// ---------------------------------------------------------------------------
__global__ void routing_r_kernel(const float* __restrict__ stats,
                                 const float* __restrict__ cw,    // (C,1,3,3)
                                 const float* __restrict__ cb,    // (C)
                                 const float* __restrict__ g, const float* __restrict__ beta,
                                 const float* __restrict__ mu, const float* __restrict__ var,
                                 float* __restrict__ rbuf) {      // (B,C)
    int idx = blockIdx.x * blockDim.x + threadIdx.x;  // b*256 + c
    if (idx >= B_ * C_) return;
    int c = idx & (C_ - 1);
    const float* s = stats + idx * 9;
    float tot = s[0], rs0 = s[1], rsN = s[2], cs0 = s[3], csN = s[4];
    float x00 = s[5], x0N = s[6], xN0 = s[7], xNN = s[8];

    float acc = 0.f;
    #pragma unroll
    for (int dy = 0; dy < 3; ++dy) {
        float re = (dy == 0) ? rsN : ((dy == 2) ? rs0 : 0.f);
        #pragma unroll
        for (int dx = 0; dx < 3; ++dx) {
            float ce = (dx == 0) ? csN : ((dx == 2) ? cs0 : 0.f);
            float corner = 0.f;
            if (dy == 0 && dx == 0) corner = xNN;
            if (dy == 0 && dx == 2) corner = xN0;
            if (dy == 2 && dx == 0) corner = x0N;
            if (dy == 2 && dx == 2) corner = x00;
            float T = tot - re - ce + corner;
            acc += cw[c * 9 + dy * 3 + dx] * T;
        }
    }
    float meanDw = acc * (1.0f / (float)HW_);
    float inv = g[c] / sqrtf(var[c] + EPS);
    float r = inv * (meanDw + cb[c]) + beta[c] - mu[c] * inv;
    rbuf[idx] = r > 0.f ? r : 0.f;
}

// ---------------------------------------------------------------------------
// Kernel C (single wave, WMMA f32 16x16x4):
//   logits(16x8) = r(16x256) @ dense_w^T  -> softmax -> wg(16x8)
//   kmix(16x2304) = wg @ kernel_embed ;  bmix(16x256) = wg @ bias_embed
// ---------------------------------------------------------------------------
__global__ void routing_mix_kernel(const float* __restrict__ rbuf,   // (16,256)
                                   const float* __restrict__ dw,     // (8,256)
                                   const float* __restrict__ db,     // (8)
                                   const float* __restrict__ ke,     // (8,2304)
                                   const float* __restrict__ be,     // (8,256)
                                   float* __restrict__ kmix,         // (16,2304)
                                   float* __restrict__ bmix) {       // (16,256)
    __shared__ float slog[16 * 16];
    __shared__ float wg[16 * E_];
    const int lane = threadIdx.x;     // one full wave -> EXEC all 1s for WMMA
    const int m    = lane & 15;       // row (A) / column (B,D)
    const int half = lane >> 4;

    // branch-free B addressing: clamp row, multiply by mask
    const int   msel  = (m < E_) ? m : 0;
    const float bmask = (m < E_) ? 1.f : 0.f;

    // ---- logits GEMM: M=16 (batch), N=16 (experts, 8 live), K=256
    v8f acc = {};
    for (int k0 = 0; k0 < C_; k0 += 4) {
        int kk = k0 + 2 * half;
        v2f a, b;
        a.x = rbuf[m * C_ + kk];
        a.y = rbuf[m * C_ + kk + 1];
        b.x = dw[msel * C_ + kk]     * bmask;
        b.y = dw[msel * C_ + kk + 1] * bmask;
        acc = __builtin_amdgcn_wmma_f32_16x16x4_f32(false, a, false, b,
                                                    (short)0, acc, false, false);
    }
    #pragma unroll
    for (int i = 0; i < 8; ++i) {
        int row = i + 8 * half;
        slog[row * 16 + m] = acc[i];
    }
    __syncthreads();

    // ---- softmax per batch row (E=8)
    if (lane < 16) {
        float l[E_];
        float mx = -1e30f;
        #pragma unroll
        for (int e = 0; e < E_; ++e) {
            l[e] = slog[lane * 16 + e] + db[e];
            mx = fmaxf(mx, l[e]);
        }
        float sum = 0.f;
        #pragma unroll
        for (int e = 0; e < E_; ++e) { l[e] = expf(l[e] - mx); sum += l[e]; }
        float isum = 1.f / sum;
        #pragma unroll
        for (int e = 0; e < E_; ++e) wg[lane * E_ + e] = l[e] * isum;
    }
    __syncthreads();

    // ---- mixing GEMMs: A = wg (16x8), two K-chunks of 4
    v2f a0, a1;
    a0.x = wg[m * E_ + 2 * half];     a0.y = wg[m * E_ + 2 * half + 1];
    a1.x = wg[m * E_ + 4 + 2 * half]; a1.y = wg[m * E_ + 4 + 2 * half + 1];

    for (int t = 0; t < NKJ / 16; ++t) {          // 144 tiles of N=16
        int n = t * 16 + m;
        v2f b0, b1;
        b0.x = ke[(2 * half) * NKJ + n];       b0.y = ke[(2 * half + 1) * NKJ + n];
        b1.x = ke[(4 + 2 * half) * NKJ + n];   b1.y = ke[(5 + 2 * half) * NKJ + n];
        v8f c = {};
        c = __builtin_amdgcn_wmma_f32_16x16x4_f32(false, a0, false, b0, (short)0, c, false, false);
        c = __builtin_amdgcn_wmma_f32_16x16x4_f32(false, a1, false, b1, (short)0, c, false, false);
        #pragma unroll
        for (int i = 0; i < 8; ++i) {
            int row = i + 8 * half;
            kmix[row * NKJ + n] = c[i];
        }
    }
    for (int t = 0; t < C_ / 16; ++t) {           // 16 tiles of N=16
        int n = t * 16 + m;
        v2f b0, b1;
        b0.x = be[(2 * half) * C_ + n];       b0.y = be[(2 * half + 1) * C_ + n];
        b1.x = be[(4 + 2 * half) * C_ + n];   b1.y = be[(5 + 2 * half) * C_ + n];
        v8f c = {};
        c = __builtin_amdgcn_wmma_f32_16x16x4_f32(false, a0, false, b0, (short)0, c, false, false);
        c = __builtin_amdgcn_wmma_f32_16x16x4_f32(false, a1, false, b1, (short)0, c, false, false);
        #pragma unroll
        for (int i = 0; i < 8; ++i) {
            int row = i + 8 * half;
            bmix[row * C_ + n] = c[i];
        }
    }
}

// ---------------------------------------------------------------------------
// Kernel D: per-sample depthwise 3x3 conv + mixed bias + BN + residual + relu.
// One block per (b,c) plane. Halo ring pre-zeroed in LDS, interior staged via
// GLOBAL_LOAD_ASYNC_TO_LDS_B32 (ASYNCcnt path) when available.
// ---------------------------------------------------------------------------
__global__ void main_conv_kernel(const float* __restrict__ x,
                                 const float* __restrict__ kmix,   // (16,2304)
                                 const float* __restrict__ bmix,   // (16,256)
                                 const float* __restrict__ g, const float* __restrict__ beta,
                                 const float* __restrict__ mu, const float* __restrict__ var,
                                 float* __restrict__ out) {
    __shared__ float tile[(H_ + 2) * TW];   // 98*98 floats = 38.4 KB
    const int bc = blockIdx.x;
    const int b  = bc >> 8;
    const int c  = bc & (C_ - 1);
    const int t  = threadIdx.x;
    const float* p = x + (size_t)bc * HW_;

    // ---- zero the halo ring: top(98) + bottom(98) + left(96) + right(96) = 388
    for (int idx = t; idx < 2 * TW + 2 * H_; idx += 256) {
        int o;
        if (idx < TW)                o = idx;                                   // top row
        else if (idx < 2 * TW)       o = (H_ + 1) * TW + (idx - TW);            // bottom row
        else if (idx < 2 * TW + H_)  o = (idx - 2 * TW + 1) * TW;               // left col
        else                         o = (idx - 2 * TW - H_ + 1) * TW + TW - 1; // right col
        tile[o] = 0.f;
    }

    // ---- stage 96x96 interior into LDS
#ifdef HAVE_ASYNC_LDS
    for (int i = 0; i < HW_ / 256; ++i) {          // 36 iterations, full EXEC
        int idx = t + i * 256;
        int h = idx / W_;
        int w = idx - h * W_;
        int o = (h + 1) * TW + (w + 1);
        __builtin_amdgcn_global_load_async_to_lds_b32(
            (gl_i32*)(p + idx), (lds_i32*)&tile[o], 0, 0);
    }
#if __has_builtin(__builtin_amdgcn_s_wait_asynccnt)
    __builtin_amdgcn_s_wait_asynccnt(0);
#else
    asm volatile("s_wait_asynccnt 0x0" ::: "memory");
#endif
#else
    for (int i = 0; i < HW_ / 256; ++i) {
        int idx = t + i * 256;
        int h = idx / W_;
        int w = idx - h * W_;
        tile[(h + 1) * TW + (w + 1)] = p[idx];
    }
#endif
    __syncthreads();

    float kw[9];
    #pragma unroll
    for (int i = 0; i < 9; ++i) kw[i] = kmix[b * NKJ + c * 9 + i];
    float bias  = bmix[b * C_ + c];
    float inv   = g[c] / sqrtf(var[c] + EPS);
    float shift = beta[c] - mu[c] * inv;

    // ---- compute 4 contiguous outputs per iteration, store as float4
    float4* out4 = (float4*)(out + (size_t)bc * HW_);
    for (int i = 0; i < HW_ / (4 * 256); ++i) {    // 9 iterations
        int q  = t + i * 256;                      // float4 index, 0..2303
        int h  = q / 24;                           // 24 float4 per row
        int wq = (q - h * 24) * 4;
        const float* tp = &tile[h * TW + wq];
        float4 r;
        float a[4];
        #pragma unroll
        for (int j = 0; j < 4; ++j) {
            float acc = tp[j]          * kw[0] + tp[j + 1]      * kw[1] + tp[j + 2]      * kw[2]
                      + tp[TW + j]     * kw[3] + tp[TW + j + 1] * kw[4] + tp[TW + j + 2] * kw[5]
                      + tp[2*TW + j]   * kw[6] + tp[2*TW + j+1] * kw[7] + tp[2*TW + j+2] * kw[8];
            float y = (acc + bias) * inv + shift + tp[TW + j + 1];   // + residual
            a[j] = y > 0.f ? y : 0.f;
        }
        r.x = a[0]; r.y = a[1]; r.z = a[2]; r.w = a[3];
        out4[q] = r;
    }
}

// ---------------------------------------------------------------------------
extern "C" void kernel_launch(void* const* d_in, const int* in_sizes, int n_in,
                              void* d_out, int out_size, void* d_ws, size_t ws_size,
                              hipStream_t stream) {
    const float* x    = (const float*)d_in[0];   // (16,256,96,96)
    const float* ke   = (const float*)d_in[1];   // (8,256,3,3)
    const float* be   = (const float*)d_in[2];   // (8,256)
    const float* cw   = (const float*)d_in[3];   // (256,1,3,3)
    const float* cb   = (const float*)d_in[4];   // (256)
    const float* cg   = (const float*)d_in[5];
    const float* cbe  = (const float*)d_in[6];
    const float* cmu  = (const float*)d_in[7];
    const float* cvar = (const float*)d_in[8];
    const float* dw   = (const float*)d_in[9];   // (8,256)
    const float* db   = (const float*)d_in[10];  // (8)
    const float* g    = (const float*)d_in[11];
    const float* bet  = (const float*)d_in[12];
    const float* mu   = (const float*)d_in[13];
    const float* var  = (const float*)d_in[14];
    float* out = (float*)d_out;

    float* ws    = (float*)d_ws;
    float* stats = ws;                        // 4096 * 9
    float* rbuf  = stats + 4096 * 9;          // 4096
    float* kmix  = rbuf + 4096;               // 16 * 2304
    float* bmix  = kmix + 16 * 2304;          // 4096

    stats_kernel<<<B_ * C_, 256, 0, stream>>>(x, stats);
    routing_r_kernel<<<(B_ * C_) / 256, 256, 0, stream>>>(stats, cw, cb, cg, cbe, cmu, cvar, rbuf);
    routing_mix_kernel<<<1, 32, 0, stream>>>(rbuf, dw, db, ke, be, kmix, bmix);
    main_conv_kernel<<<B_ * C_, 256, 0, stream>>>(x, kmix, bmix, g, bet, mu, var, out);
}